// Model_44212393345516
// MI455X (gfx1250) — compile-verified
//
#include <hip/hip_runtime.h>
#include <stdint.h>

#define HD    64
#define LBACK 32
#define PWIN  16
#define WLEN  48
#define LSTRF 68    // row stride (floats) for sF/sM: conflict-free col-pair b64 reads
#define PSTR  164   // pair-row stride (floats) for sP: (2 rows interleaved) + pad

typedef __attribute__((ext_vector_type(2))) float v2f;
typedef __attribute__((ext_vector_type(8))) float v8f;

// P element (r,c) in pair-interleaved LDS layout
#define PIDX(r, c) ((((r) >> 1) * PSTR) + 2 * (c) + ((r) & 1))

__device__ __forceinline__ v8f wmma4(v2f a, v2f b, v8f c) {
  // D = A(16x4) * B(4x16) + C, full f32
  return __builtin_amdgcn_wmma_f32_16x16x4_f32(false, a, false, b, (short)0, c, false, false);
}

__device__ __forceinline__ v2f lds2(const float* p) { return *(const v2f*)p; }

__global__ __launch_bounds__(128) void kalman_kernel(
    const float* __restrict__ x,  const float* __restrict__ Fw,
    const float* __restrict__ Fb, const float* __restrict__ Hw,
    const float* __restrict__ Hb, const float* __restrict__ s0,
    const float* __restrict__ P0, const float* __restrict__ Q,
    const float* __restrict__ R,  float* __restrict__ out)
{
  __shared__ float sP[32 * PSTR];    // covariance, pair-interleaved (persistent)
  __shared__ float sM[HD * LSTRF];   // M = F @ P, row-major
  __shared__ float sF[HD * LSTRF];   // per-step F, row-major
  __shared__ float s_h[HD], s_hs[HD], s_hsp[HD], s_u[HD], s_v[HD], s_fb[HD];
  __shared__ float s_red[128];
  __shared__ float s_scal[2];

  const int tid  = threadIdx.x;
  const int b    = blockIdx.x;
  const int wave = tid >> 5;
  const int lane = tid & 31;
  const int l15  = lane & 15;
  const int hi   = lane >> 4;   // half-wave select (K/M high split)
  const int row0 = wave * 16;   // this wave's output tile-row

  // ---- init: P0 -> LDS (pair layout), hs -> LDS, Q fragments -> registers ----
  for (int i = tid; i < HD * HD; i += 128)
    sP[PIDX(i >> 6, i & 63)] = P0[i];
  if (tid < HD) s_hs[tid] = s0[tid];

  v8f qf[4];
  #pragma unroll
  for (int j = 0; j < 4; ++j)
    #pragma unroll
    for (int r = 0; r < 8; ++r)
      qf[j][r] = Q[(row0 + r + 8 * hi) * HD + 16 * j + l15];

  __syncthreads();

  // ================= Kalman filtering: 32 sequential steps =================
  for (int t = 0; t < LBACK; ++t) {
    // ---- stage F[t] into sF via async copy (8 x b128 per thread) ----
    {
      const float* gsrc = Fw + t * HD * HD;
      #pragma unroll
      for (int n = 0; n < 8; ++n) {
        const int f    = tid + 128 * n;        // float4 index 0..1023
        const int flat = f * 4;
        const int r = flat >> 6, c = flat & 63;
        unsigned lds_off = (unsigned)(size_t)(&sF[r * LSTRF + c]);
        uint64_t gaddr   = (uint64_t)(size_t)(gsrc + flat);
        asm volatile("global_load_async_to_lds_b128 %0, %1, off"
                     :: "v"(lds_off), "v"(gaddr) : "memory");
      }
    }
    if (tid < HD) { s_h[tid] = Hw[t * HD + tid]; s_fb[tid] = Fb[t * HD + tid]; }
    if (t + 1 < LBACK)
      __builtin_prefetch(Fw + (t + 1) * HD * HD + tid * 32, 0, 0);
    asm volatile("s_wait_asynccnt 0x0" ::: "memory");
    __syncthreads();

    // ---- M = F @ P : wave computes rows [row0, row0+16) ----
    v2f af[16];
    #pragma unroll
    for (int k = 0; k < 16; ++k)             // A-frag: adjacent cols of F row (b64)
      af[k] = lds2(&sF[(row0 + l15) * LSTRF + 4 * k + 2 * hi]);
    #pragma unroll
    for (int j = 0; j < 4; ++j) {
      v8f acc = {};
      #pragma unroll
      for (int k = 0; k < 16; ++k) {         // B-frag: interleaved row-pair of P (b64)
        v2f bp = lds2(&sP[(2 * k + hi) * PSTR + 2 * (16 * j + l15)]);
        acc = wmma4(af[k], bp, acc);
      }
      #pragma unroll
      for (int r = 0; r < 8; ++r)
        sM[(row0 + r + 8 * hi) * LSTRF + 16 * j + l15] = acc[r];
    }
    // no barrier: wave reads back only the M rows it wrote (LDS in-order per wave)

    // ---- P_pred = M @ F^T + Q  (acc seeded from register-resident Q frags) ----
    #pragma unroll
    for (int k = 0; k < 16; ++k)
      af[k] = lds2(&sM[(row0 + l15) * LSTRF + 4 * k + 2 * hi]);
    #pragma unroll
    for (int j = 0; j < 4; ++j) {
      v8f acc = qf[j];
      #pragma unroll
      for (int k = 0; k < 16; ++k) {         // B-frag of F^T: adjacent cols of F row (b64)
        v2f bf = lds2(&sF[(16 * j + l15) * LSTRF + 4 * k + 2 * hi]);
        acc = wmma4(af[k], bf, acc);
      }
      #pragma unroll
      for (int r = 0; r < 8; r += 2) {       // paired C-store into P pair layout (b64)
        v2f st; st.x = acc[r]; st.y = acc[r + 1];
        *(v2f*)&sP[((row0 + r + 8 * hi) >> 1) * PSTR + 2 * (16 * j + l15)] = st;
      }
    }
    __syncthreads();

    // ---- hs_pred = F @ hs + Fb ----
    if (tid < HD) {
      float a = s_fb[tid];
      for (int k = 0; k < HD; ++k) a += sF[tid * LSTRF + k] * s_hs[k];
      s_hsp[tid] = a;
    }
    __syncthreads();

    // ---- u = P_pred h ; v = h^T P_pred  (TC==1 -> rank-1 update) ----
    if (tid < HD) {
      float a = 0.f;
      for (int k = 0; k < HD; ++k) a += sP[PIDX(tid, k)] * s_h[k];
      s_u[tid] = a;
    } else {
      const int j = tid - HD;
      float a = 0.f;
      for (int k = 0; k < HD; ++k) a += s_h[k] * sP[PIDX(k, j)];
      s_v[j] = a;
    }
    __syncthreads();

    // ---- S = h.u + R ; innovation = y - (h.hs_pred + Hb)  (two parallel trees) ----
    s_red[tid] = (tid < HD) ? s_h[tid] * s_u[tid]
                            : s_h[tid - HD] * s_hsp[tid - HD];
    __syncthreads();
    #pragma unroll
    for (int sh = 32; sh >= 1; sh >>= 1) {
      if ((tid & 63) < sh) s_red[tid] += s_red[tid + sh];
      __syncthreads();
    }
    if (tid == 0) {
      float Sv  = s_red[0] + R[0];
      s_scal[0] = 1.0f / Sv;
      s_scal[1] = x[b * LBACK + t] - (s_red[64] + Hb[t]);
    }
    __syncthreads();
    const float invS  = s_scal[0];
    const float innov = s_scal[1];

    if (tid < HD) s_hs[tid] = s_hsp[tid] + invS * innov * s_u[tid];
    for (int i = tid; i < 32 * HD; i += 128) {   // P <- P_pred - (u v^T)/S, paired rows
      const int rp = i >> 6, c = i & 63;
      v2f pv = *(v2f*)&sP[rp * PSTR + 2 * c];
      const float s = s_v[c] * invS;
      pv.x -= s_u[2 * rp + 0] * s;
      pv.y -= s_u[2 * rp + 1] * s;
      *(v2f*)&sP[rp * PSTR + 2 * c] = pv;
    }
    __syncthreads();
  }

  // ================= prediction: 16 cheap matvec steps =================
  for (int t = LBACK; t < WLEN; ++t) {
    const float* Ft = Fw + t * HD * HD;
    if (tid < HD) {
      float a = Fb[t * HD + tid];
      for (int k = 0; k < HD; ++k) a += Ft[tid * HD + k] * s_hs[k];
      s_hsp[tid] = a;
    }
    __syncthreads();
    if (tid < HD) {
      s_hs[tid]  = s_hsp[tid];
      s_red[tid] = Hw[t * HD + tid] * s_hsp[tid];
    }
    __syncthreads();
    #pragma unroll
    for (int sh = 32; sh >= 1; sh >>= 1) {
      if (tid < sh) s_red[tid] += s_red[tid + sh];
      __syncthreads();
    }
    if (tid == 0) out[b * PWIN + (t - LBACK)] = s_red[0] + Hb[t];
    __syncthreads();
  }
}

extern "C" void kernel_launch(void* const* d_in, const int* in_sizes, int n_in,
                              void* d_out, int out_size, void* d_ws, size_t ws_size,
                              hipStream_t stream) {
  const float* x  = (const float*)d_in[0];
  const float* Fw = (const float*)d_in[1];
  const float* Fb = (const float*)d_in[2];
  const float* Hw = (const float*)d_in[3];
  const float* Hb = (const float*)d_in[4];
  const float* s0 = (const float*)d_in[5];
  const float* P0 = (const float*)d_in[6];
  const float* Q  = (const float*)d_in[7];
  const float* R  = (const float*)d_in[8];
  float* out = (float*)d_out;

  hipLaunchKernelGGL(kalman_kernel, dim3(4096), dim3(128), 0, stream,
                     x, Fw, Fb, Hw, Hb, s0, P0, Q, R, out);
}